// GNNRoutingModel_5884105195871
// MI455X (gfx1250) — compile-verified
//
#include <hip/hip_runtime.h>
#include <hip/hip_bf16.h>

typedef __attribute__((ext_vector_type(2))) float v2f;
typedef __attribute__((ext_vector_type(8))) float v8f;

#define NODE_DIM 64

// ---------------- degree / normalization ----------------
__global__ void k_deg_init(float* __restrict__ deg, int n) {
    int i = blockIdx.x * blockDim.x + threadIdx.x;
    if (i < n) deg[i] = 1.0f;  // self-loop
}

__global__ void k_deg_count(float* __restrict__ deg, const int* __restrict__ dst, int e) {
    int i = blockIdx.x * blockDim.x + threadIdx.x;
    if (i < e) atomicAdd(&deg[dst[i]], 1.0f);
}

__global__ void k_dinv(float* __restrict__ deg, int n) {
    int i = blockIdx.x * blockDim.x + threadIdx.x;
    if (i < n) deg[i] = rsqrtf(deg[i]);  // deg >= 1 always
}

// ---------------- dense GEMM: H[N,64] = act(A[N,64]) @ W[64,64] ----------------
// One 16-row tile per block; 4 waves, each owns one 16-col tile.
// Uses V_WMMA_F32_16X16X4_F32 (fp32 exact), 16 MACs over K=64.
// RELU is a compile-time template arg (no per-element cndmask in the hot loop).
// Out-of-range A rows are only pointer-clamped: a bogus A row m corrupts only
// D row m, which the guarded store never writes.
template <bool RELU>
__global__ void k_gemm64_wmma(const float* __restrict__ A, const float* __restrict__ W,
                              float* __restrict__ H, int n) {
    const int rowBase = blockIdx.x * 16;
    const int lane    = threadIdx.x & 31;
    const int colBase = (threadIdx.x >> 5) * 16;   // wave -> col tile
    const int m       = lane & 15;
    const int khalf   = (lane >> 4) * 2;           // 0 or 2 (A/B K-split across half-waves)

    const int   arowIdx = rowBase + m;
    const float* arow   = A + (size_t)((arowIdx < n) ? arowIdx : (n - 1)) * NODE_DIM;
    const int   nc      = colBase + m;

    v8f c = {};
#pragma unroll
    for (int k0 = 0; k0 < NODE_DIM; k0 += 4) {
        // A: 16x4 fp32 tile (ISA 7.12.2 layout)
        v2f a;
        a.x = arow[k0 + khalf + 0];
        a.y = arow[k0 + khalf + 1];
        if (RELU) { a.x = fmaxf(a.x, 0.0f); a.y = fmaxf(a.y, 0.0f); }
        // B: 4x16 fp32 tile (K rows striped like A, N across lanes)
        v2f b;
        b.x = W[(k0 + khalf + 0) * NODE_DIM + nc];
        b.y = W[(k0 + khalf + 1) * NODE_DIM + nc];
        c = __builtin_amdgcn_wmma_f32_16x16x4_f32(false, a, false, b, (short)0, c, false, false);
    }

    // D: 16x16 f32, VGPR r -> row r + (lane>>4)*8, col lane&15
#pragma unroll
    for (int r = 0; r < 8; ++r) {
        const int row = rowBase + r + (lane >> 4) * 8;
        if (row < n) H[(size_t)row * NODE_DIM + colBase + (lane & 15)] = c[r];
    }
}

// ---------------- aggregation init: agg = h*dinv^2 + bias (self-loop + bias) ----------------
__global__ void k_agg_init(float* __restrict__ agg, const float* __restrict__ h,
                           const float* __restrict__ dinv, const float* __restrict__ bias, int n) {
    int i = blockIdx.x * blockDim.x + threadIdx.x;
    if (i < n * NODE_DIM) {
        int node = i >> 6, f = i & 63;
        float di = dinv[node];
        agg[i] = h[i] * di * di + bias[f];
    }
}

// ---------------- edge scatter: agg[dst] += h[src] * dinv[src]*dinv[dst] ----------------
// 16 threads per edge, float4 per thread.
__global__ void k_scatter(float* __restrict__ agg, const float* __restrict__ h,
                          const float* __restrict__ dinv,
                          const int* __restrict__ src, const int* __restrict__ dst, int e) {
    int i = blockIdx.x * blockDim.x + threadIdx.x;
    if (i >= e * 16) return;
    int edge = i >> 4;
    int f    = (i & 15) * 4;
    int s = src[edge], d = dst[edge];
    float norm = dinv[s] * dinv[d];
    const float4 v = *(const float4*)(h + (size_t)s * NODE_DIM + f);
    float* o = agg + (size_t)d * NODE_DIM + f;
    atomicAdd(o + 0, v.x * norm);
    atomicAdd(o + 1, v.y * norm);
    atomicAdd(o + 2, v.z * norm);
    atomicAdd(o + 3, v.w * norm);
}

// ---------------- edge scorer: one wave32 per edge ----------------
// score = dot(relu(h[src]), We[0:64]) + dot(relu(h[dst]), We[64:128]) + dot(ea, We[128:144]) + be
__global__ void k_edge_score(float* __restrict__ out, const float* __restrict__ h,
                             const float* __restrict__ ea, const float* __restrict__ We,
                             const float* __restrict__ be,
                             const int* __restrict__ src, const int* __restrict__ dst, int e) {
    const int lane = threadIdx.x & 31;
    const int edge = blockIdx.x * 8 + (threadIdx.x >> 5);
    if (edge >= e) return;
    const int s = src[edge], d = dst[edge];

    const v2f hs = *(const v2f*)(h + (size_t)s * NODE_DIM + 2 * lane);
    const v2f hd = *(const v2f*)(h + (size_t)d * NODE_DIM + 2 * lane);
    const v2f ws = *(const v2f*)(We + 2 * lane);
    const v2f wd = *(const v2f*)(We + 64 + 2 * lane);

    float p = fmaxf(hs.x, 0.0f) * ws.x + fmaxf(hs.y, 0.0f) * ws.y
            + fmaxf(hd.x, 0.0f) * wd.x + fmaxf(hd.y, 0.0f) * wd.y;
    if (lane < 16) p += ea[(size_t)edge * 16 + lane] * We[128 + lane];

#pragma unroll
    for (int off = 16; off >= 1; off >>= 1) p += __shfl_xor(p, off, 32);

    if (lane == 0) out[edge] = p + be[0];
}

// ---------------- host launch ----------------
extern "C" void kernel_launch(void* const* d_in, const int* in_sizes, int n_in,
                              void* d_out, int out_size, void* d_ws, size_t ws_size,
                              hipStream_t stream) {
    const float* x  = (const float*)d_in[0];
    const int*   ei = (const int*)d_in[1];
    const float* ea = (const float*)d_in[2];
    const float* W1 = (const float*)d_in[3];
    const float* b1 = (const float*)d_in[4];
    const float* W2 = (const float*)d_in[5];
    const float* b2 = (const float*)d_in[6];
    const float* We = (const float*)d_in[7];
    const float* be = (const float*)d_in[8];
    float* out = (float*)d_out;

    const int N = in_sizes[0] / NODE_DIM;
    const int E = in_sizes[1] / 2;
    const int* src = ei;
    const int* dst = ei + E;

    // workspace layout: dinv | h (h1 then h2) | agg (agg1 then agg2)
    char* ws = (char*)d_ws;
    const size_t dinvBytes = (((size_t)N * 4) + 511) & ~(size_t)511;
    const size_t hBytes    = (size_t)N * NODE_DIM * 4;
    float* dinv = (float*)ws;
    float* h    = (float*)(ws + dinvBytes);
    float* agg  = (float*)(ws + dinvBytes + hBytes);

    const int TB = 256;
    const int gN    = (N + TB - 1) / TB;
    const int gE    = (E + TB - 1) / TB;
    const int gNF   = (N * NODE_DIM + TB - 1) / TB;
    const int gSc   = (E * 16 + TB - 1) / TB;
    const int gGemm = (N + 15) / 16;       // 16-row tiles, 128-thread blocks (4 waves = 4 col tiles)
    const int gEdge = (E + 7) / 8;         // 8 edges per 256-thread block

    // normalization
    k_deg_init <<<gN, TB, 0, stream>>>(dinv, N);
    k_deg_count<<<gE, TB, 0, stream>>>(dinv, dst, E);
    k_dinv     <<<gN, TB, 0, stream>>>(dinv, N);

    // layer 1: h1 = x@W1 ; agg1 = scatter + selfloop + b1
    k_gemm64_wmma<false><<<gGemm, 128, 0, stream>>>(x, W1, h, N);
    k_agg_init          <<<gNF, TB, 0, stream>>>(agg, h, dinv, b1, N);
    k_scatter           <<<gSc, TB, 0, stream>>>(agg, h, dinv, src, dst, E);

    // layer 2: h2 = relu(agg1)@W2 ; agg2 = scatter + selfloop + b2 (buffers reused, stream-ordered)
    k_gemm64_wmma<true><<<gGemm, 128, 0, stream>>>(agg, W2, h, N);
    k_agg_init         <<<gNF, TB, 0, stream>>>(agg, h, dinv, b2, N);
    k_scatter          <<<gSc, TB, 0, stream>>>(agg, h, dinv, src, dst, E);

    // edge scoring on relu(agg2)
    k_edge_score<<<gEdge, TB, 0, stream>>>(out, agg, ea, We, be, src, dst, E);
}